// BiGRU_37984690765884
// MI455X (gfx1250) — compile-verified
//
#include <hip/hip_runtime.h>
#include <hip/hip_bf16.h>

// BiGRU: 3 layers, bidirectional, H=23, B=2048, T=512, F_IN=32, OUT=8.
//  - normalize_kernel: x -> bufA [T,B,48] (cols 32..47 zeroed)
//  - gru_dir_kernel<IN>: fused (input GEMM + recurrence) per (layer,dir).
//      grid = (B/32, 2 dirs); block = 160 threads = 5 wave32s.
//      Wave w owns N-tile w of the 69 gate columns for two 16-row M-tiles.
//      W_ih/W_hh WMMA B-fragments + biases preloaded into registers once.
//      Input tiles are double-buffered in LDS and filled with
//      GLOBAL_LOAD_ASYNC_TO_LDS_B128 (ASYNCcnt) one step ahead, so HBM
//      latency is off the sequential recurrence critical path.
//  - fc_kernel: out = last @ fc_w^T + fc_b, de-normalized.
// Workspace: two ping-pong buffers of [512,2048,48] f32 = 2 * 192 MB.

#define H_ 23
#define G_ 69       // 3*H
#define B_ 2048
#define T_ 512
#define STRIDE_ 48  // feature stride in seq buffers (192 B, 16B-aligned rows)
#define XS_ 52      // x_lds row stride in floats (208 B: 16B-aligned, conflict-free)

typedef float v2f __attribute__((ext_vector_type(2)));
typedef float v8f __attribute__((ext_vector_type(8)));

__device__ __forceinline__ v8f splat8(float v) {
    v8f r;
#pragma unroll
    for (int i = 0; i < 8; ++i) r[i] = v;
    return r;
}

__device__ __forceinline__ float fsig(float x) {
    return __builtin_amdgcn_rcpf(1.0f + __expf(-x));
}
__device__ __forceinline__ float ftanh(float x) {
    // 1 - 2/(e^{2x}+1): saturates to +/-1 as e^{2x} -> inf/0
    return 1.0f - 2.0f * __builtin_amdgcn_rcpf(__expf(2.0f * x) + 1.0f);
}

// Issue 1536 contiguous bytes (12 floats into each of 32 rows) as 96 lanes x b128 async
// copies into LDS at byte offset lds_base (row stride 208 B).
__device__ __forceinline__ void issue_tile_async(const float* gsrc, unsigned lds_base,
                                                 int tid) {
    if (tid < 96) {
        unsigned ldsb = lds_base + (unsigned)(tid / 3) * (XS_ * 4) + (unsigned)(tid % 3) * 16;
        unsigned goff = (unsigned)tid * 16;
        unsigned long long base = (unsigned long long)(uintptr_t)gsrc;
        asm volatile("global_load_async_to_lds_b128 %0, %1, %2"
                     :
                     : "v"(ldsb), "v"(goff), "s"(base)
                     : "memory");
    }
}

__global__ void normalize_kernel(const float* __restrict__ x,
                                 const float* __restrict__ fmean,
                                 const float* __restrict__ fstd,
                                 float* __restrict__ bufA) {
    size_t idx = (size_t)blockIdx.x * 256 + threadIdx.x;
    const size_t total = (size_t)T_ * B_ * STRIDE_;
    if (idx >= total) return;
    int c = (int)(idx % STRIDE_);
    size_t bt = idx / STRIDE_;
    int b = (int)(bt % B_);
    int t = (int)(bt / B_);
    float v = 0.0f;
    if (c < 32) {
        float s = fstd[c];
        s = (s == 0.0f) ? 1.0f : s;
        if (s != 1.0f) {
            v = (x[((size_t)b * T_ + t) * 32 + c] - fmean[c]) / s;
        }
    }
    bufA[idx] = v;
}

template <int IN>
__global__ __launch_bounds__(160)
void gru_dir_kernel(const float* __restrict__ seq_in, float* __restrict__ seq_out,
                    const float* __restrict__ w_ih_all, const float* __restrict__ w_hh_all,
                    const float* __restrict__ b_ih_all, const float* __restrict__ b_hh_all) {
    constexpr int KI = (IN + 3) / 4;   // input-GEMM K-steps (8 or 12)
    const int dir  = blockIdx.y;       // 0 = fwd, 1 = bwd
    const int b0   = blockIdx.x * 32;  // batch-chunk base
    const int tid  = threadIdx.x;
    const int wave = tid >> 5;         // 0..4 -> N-tile
    const int lane = tid & 31;
    const int lo   = lane & 15;
    const int hi   = lane >> 4;

    const float* w_ih = w_ih_all + dir * G_ * IN;
    const float* w_hh = w_hh_all + dir * G_ * H_;
    const float* b_ih = b_ih_all + dir * G_;
    const float* b_hh = b_hh_all + dir * G_;

    __shared__ __align__(16) float x_lds[2][32 * XS_];  // double-buffered input tiles
    __shared__ float h_lds[32 * 25];   // hidden state, stride 25 (col 23 = K pad, stays 0)
    __shared__ float xg_lds[32 * 81];  // input-side gates, stride 81
    __shared__ float hg_lds[32 * 81];  // hidden-side gates

    for (int i = tid; i < 32 * 25; i += 160) h_lds[i] = 0.0f;

    // ---- Preload timestep-invariant B-fragments + biases into registers ----
    // B-fragment layout (16x16x4 f32): VGPR0: K=4k+0 (lanes 0-15) / K=4k+2 (16-31),
    //                                  VGPR1: K=4k+1            / K=4k+3.
    const int ncol = wave * 16 + lo;        // gate column 0..79
    const bool nok = (ncol < G_);
    v2f whhB[6];
#pragma unroll
    for (int k = 0; k < 6; ++k) {
        int k0 = 4 * k + 2 * hi;
        float a = (nok && k0 < H_)     ? w_hh[ncol * H_ + k0]     : 0.0f;
        float b = (nok && k0 + 1 < H_) ? w_hh[ncol * H_ + k0 + 1] : 0.0f;
        whhB[k] = (v2f){a, b};
    }
    v2f wihB[KI];
#pragma unroll
    for (int k = 0; k < KI; ++k) {
        int k0 = 4 * k + 2 * hi;
        float a = (nok && k0 < IN)     ? w_ih[ncol * IN + k0]     : 0.0f;
        float b = (nok && k0 + 1 < IN) ? w_ih[ncol * IN + k0 + 1] : 0.0f;
        wihB[k] = (v2f){a, b};
    }
    const float bih = nok ? b_ih[ncol] : 0.0f;
    const float bhh = nok ? b_hh[ncol] : 0.0f;
    const int  out_off  = dir * H_;
    const bool reversed = (dir == 1);

    const unsigned xbase0 = (unsigned)(uintptr_t)&x_lds[0][0];
    const unsigned xbase1 = (unsigned)(uintptr_t)&x_lds[1][0];

    // prefetch first tile into buffer 0
    {
        const int t0 = reversed ? (T_ - 1) : 0;
        issue_tile_async(seq_in + ((size_t)t0 * B_ + b0) * STRIDE_, xbase0, tid);
    }

    for (int ts = 0; ts < T_; ++ts) {
        const int t   = reversed ? (T_ - 1 - ts) : ts;
        const int cur = ts & 1;

        asm volatile("s_wait_asynccnt 0" ::: "memory");
        __syncthreads();  // tile[cur] ready for all waves; prev gate phase done

        if (ts + 1 < T_) {  // kick off DMA for next step's tile into the other buffer
            const int tn = reversed ? (T_ - 2 - ts) : (ts + 1);
            issue_tile_async(seq_in + ((size_t)tn * B_ + b0) * STRIDE_,
                             cur ? xbase0 : xbase1, tid);
        }

        // ---- WMMA: hg = h @ Whh^T + bhh ; xg = x @ Wih^T + bih ----
        const float* xc = &x_lds[cur][0];
        v8f acch[2], accx[2];
#pragma unroll
        for (int mt = 0; mt < 2; ++mt) { acch[mt] = splat8(bhh); accx[mt] = splat8(bih); }

#pragma unroll
        for (int k = 0; k < 6; ++k) {
            const int k0 = 4 * k + 2 * hi;
#pragma unroll
            for (int mt = 0; mt < 2; ++mt) {
                const int row = mt * 16 + lo;
                v2f a = (v2f){h_lds[row * 25 + k0], h_lds[row * 25 + k0 + 1]};
                acch[mt] = __builtin_amdgcn_wmma_f32_16x16x4_f32(
                    false, a, false, whhB[k], (short)0, acch[mt], false, false);
            }
        }
#pragma unroll
        for (int k = 0; k < KI; ++k) {
            const int k0 = 4 * k + 2 * hi;
#pragma unroll
            for (int mt = 0; mt < 2; ++mt) {
                const int row = mt * 16 + lo;
                v2f a = (v2f){xc[row * XS_ + k0], xc[row * XS_ + k0 + 1]};
                accx[mt] = __builtin_amdgcn_wmma_f32_16x16x4_f32(
                    false, a, false, wihB[k], (short)0, accx[mt], false, false);
            }
        }

        // C/D layout: VGPR v -> M = v + 8*hi (+16*mt), N = lo
#pragma unroll
        for (int mt = 0; mt < 2; ++mt) {
#pragma unroll
            for (int v = 0; v < 8; ++v) {
                const int row = mt * 16 + v + 8 * hi;
                xg_lds[row * 81 + ncol] = accx[mt][v];
                hg_lds[row * 81 + ncol] = acch[mt][v];
            }
        }
        __syncthreads();

        // ---- gates + state update + output store ----
        for (int i = tid; i < 32 * H_; i += 160) {
            const int m = i / H_, j = i % H_;
            const float xr = xg_lds[m * 81 + j],        hr = hg_lds[m * 81 + j];
            const float xz = xg_lds[m * 81 + j + H_],   hz = hg_lds[m * 81 + j + H_];
            const float xn = xg_lds[m * 81 + j + 2*H_], hn = hg_lds[m * 81 + j + 2*H_];
            const float r = fsig(xr + hr);
            const float z = fsig(xz + hz);
            const float n = ftanh(xn + r * hn);
            const float hnew = (1.0f - z) * n + z * h_lds[m * 25 + j];
            h_lds[m * 25 + j] = hnew;
            seq_out[((size_t)t * B_ + b0 + m) * STRIDE_ + out_off + j] = hnew;
        }
        if (dir == 0 && tid < 32) {  // keep K-pad columns of next layer's input zero
            float* dst = seq_out + ((size_t)t * B_ + b0 + tid) * STRIDE_;
            dst[46] = 0.0f;
            dst[47] = 0.0f;
        }
        // next-iteration loop-top barrier separates these LDS writes from reuse
    }
}

__global__ void fc_kernel(const float* __restrict__ seq_last,
                          const float* __restrict__ fc_w, const float* __restrict__ fc_b,
                          const float* __restrict__ omean, const float* __restrict__ ostd,
                          float* __restrict__ out) {
    int idx = blockIdx.x * 256 + threadIdx.x;  // B*OUT = 16384
    if (idx >= B_ * 8) return;
    const int b = idx / 8, o = idx % 8;
    const float* last = seq_last + ((size_t)(T_ - 1) * B_ + b) * STRIDE_;
    float acc = fc_b[o];
#pragma unroll
    for (int j = 0; j < 2 * H_; ++j) acc += last[j] * fc_w[o * (2 * H_) + j];
    out[idx] = acc * ostd[o] + omean[o];
}

extern "C" void kernel_launch(void* const* d_in, const int* in_sizes, int n_in,
                              void* d_out, int out_size, void* d_ws, size_t ws_size,
                              hipStream_t stream) {
    const float* x          = (const float*)d_in[0];
    const float* fmean      = (const float*)d_in[1];
    const float* fstd       = (const float*)d_in[2];
    const float* omean      = (const float*)d_in[3];
    const float* ostd       = (const float*)d_in[4];
    const float* w_ih_l0    = (const float*)d_in[5];   // [2,69,32]
    const float* w_hh_l0    = (const float*)d_in[6];   // [2,69,23]
    const float* b_ih_l0    = (const float*)d_in[7];   // [2,69]
    const float* b_hh_l0    = (const float*)d_in[8];   // [2,69]
    const float* w_ih_rest  = (const float*)d_in[9];   // [2,2,69,46]
    const float* w_hh_rest  = (const float*)d_in[10];  // [2,2,69,23]
    const float* b_ih_rest  = (const float*)d_in[11];  // [2,2,69]
    const float* b_hh_rest  = (const float*)d_in[12];  // [2,2,69]
    const float* fc_w       = (const float*)d_in[13];  // [8,46]
    const float* fc_b       = (const float*)d_in[14];  // [8]
    float* out = (float*)d_out;

    const size_t buf_elems = (size_t)T_ * B_ * STRIDE_;  // 50,331,648 f32
    float* bufA = (float*)d_ws;                           // layer0 in  (x_norm)
    float* bufB = bufA + buf_elems;                       // ping-pong

    // normalize x -> bufA (also zeroes pad columns 32..47)
    {
        const size_t total = buf_elems;
        const int grid = (int)((total + 255) / 256);
        normalize_kernel<<<grid, 256, 0, stream>>>(x, fmean, fstd, bufA);
    }

    dim3 ggrid(B_ / 32, 2);
    // layer 0: bufA -> bufB
    gru_dir_kernel<32><<<ggrid, 160, 0, stream>>>(bufA, bufB, w_ih_l0, w_hh_l0,
                                                  b_ih_l0, b_hh_l0);
    // layer 1: bufB -> bufA
    gru_dir_kernel<46><<<ggrid, 160, 0, stream>>>(bufB, bufA, w_ih_rest, w_hh_rest,
                                                  b_ih_rest, b_hh_rest);
    // layer 2: bufA -> bufB
    gru_dir_kernel<46><<<ggrid, 160, 0, stream>>>(
        bufA, bufB,
        w_ih_rest + (size_t)2 * G_ * 46, w_hh_rest + (size_t)2 * G_ * H_,
        b_ih_rest + 2 * G_,              b_hh_rest + 2 * G_);

    // final FC from bufB[T-1]
    fc_kernel<<<(B_ * 8 + 255) / 256, 256, 0, stream>>>(bufB, fc_w, fc_b, omean, ostd, out);
}